// MEFSSIM_24876450579159
// MI455X (gfx1250) — compile-verified
//
#include <hip/hip_runtime.h>

// MEF-SSIM for MI455X (gfx1250, wave32).
// Vertical Gaussian blur via V_WMMA_F32_16X16X4_F32 (full f32 precision --
// sigma = blur(x^2) - mu^2 cancels catastrophically in reduced precision).
// Assumed WMMA f32 16x16x4 operand layouts (CDNA5 ISA 7.12.2):
//   A (16x4): lane L holds row M=L&15; VGPR0 = K {0 | 2}, VGPR1 = K {1 | 3}
//             for lane-half {0-15 | 16-31}.
//   B (4x16): mirrored half-split: lane L holds col N=L&15; VGPR0 = K {0 | 2},
//             VGPR1 = K {1 | 3}.
//   C/D: VGPR i: lanes0-15 -> (M=i, N=lane); lanes16-31 -> (M=i+8, N=lane-16).

typedef __attribute__((ext_vector_type(2))) float v2f;
typedef __attribute__((ext_vector_type(8))) float v8f;

#define H_IMG 1024
#define W_IMG 1024
#define K_EXP 8
#define NPIX (H_IMG * W_IMG)
#define TILES_X 64
#define NTILES (TILES_X * TILES_X)
#define REG 26     // 16 + 2*5 halo
#define RSTR 27    // padded LDS row stride
#define WAVES 8
#define C1_V 0.0001f   // 0.01^2
#define C2_V 0.0009f   // 0.03^2
#define DENOM 0.08f    // 2*0.2^2

// Separable blur of one raw 26x26 LDS tile -> 16x16 blurred tile in WMMA D
// layout. acoef = per-lane slice of the Gaussian band matrix G (hoisted,
// loop-invariant: avoids dynamic-register-index select chains per map).
__device__ __forceinline__ v8f blur_tile(const float* raw, const float* gw,
                                         const v2f* acoef, int n, int kb) {
  // Horizontal blur: lane computes H[row][n] for the 14 rows it feeds into B.
  float hv[14];
#pragma unroll
  for (int c = 0; c < 7; ++c) {
#pragma unroll
    for (int t = 0; t < 2; ++t) {
      int row = 4 * c + kb + t;
      float acc = 0.0f;
      if (row < REG) {
#pragma unroll
        for (int d = 0; d < 11; ++d)
          acc += gw[d] * raw[row * RSTR + n + d];
      }
      hv[2 * c + t] = acc;
    }
  }

  // Vertical blur: Out = G(16x28) x H(28x16) as 7 chained K=4 f32 WMMAs.
  v8f acc = {0.f, 0.f, 0.f, 0.f, 0.f, 0.f, 0.f, 0.f};
#pragma unroll
  for (int c = 0; c < 7; ++c) {
    v2f b;
    b.x = hv[2 * c];
    b.y = hv[2 * c + 1];
    acc = __builtin_amdgcn_wmma_f32_16x16x4_f32(false, acoef[c], false, b,
                                                (short)0, acc, false, false);
  }
  return acc;
}

__global__ __launch_bounds__(256) void mef_phase1(
    const float* __restrict__ out_img,   // [3,H,W]
    const float* __restrict__ bunch,     // [8,3,H,W]
    const float* __restrict__ window,    // [1,3,11,11]
    float* __restrict__ muY_maps,        // ws: [8,H,W]
    float* __restrict__ muX_map,         // ws: [H,W]
    float* __restrict__ cs_map,          // ws: [H,W]
    float* __restrict__ tileSums,        // ws: [8,NTILES]
    int* __restrict__ patch_index)       // d_out+1: [H,W]
{
  __shared__ float s_rawA[WAVES][REG * RSTR];
  __shared__ float s_rawB[WAVES][REG * RSTR];
  __shared__ float s_rawC[WAVES][REG * RSTR];
  __shared__ float s_gw[11];             // for dynamic-index coef build

  const int tid  = threadIdx.x;
  const int lane = tid & 31;
  const int wv   = tid >> 5;
  const int tile = blockIdx.x * WAVES + wv;
  const int ty = tile / TILES_X, tx = tile % TILES_X;
  const int row0 = ty * 16, col0 = tx * 16;

  const int n  = lane & 15;        // B column / output column
  const int kb = (lane >> 4) * 2;  // K sub-rows this lane supplies
  const int M  = lane & 15;        // A row / output row (within half)

  // Recover exact normalized 1D gaussian from the window input:
  // window[0][0][5][j] = g[5]*g[j]/3  ->  g[j] = win55[j] / sum_j(win55[j]).
  float gw[11];
  {
    float s = 0.0f;
#pragma unroll
    for (int j = 0; j < 11; ++j) { gw[j] = window[55 + j]; s += gw[j]; }
    float inv = 1.0f / s;
#pragma unroll
    for (int j = 0; j < 11; ++j) gw[j] *= inv;
  }
  if (tid < 11) s_gw[tid] = gw[tid];
  __syncthreads();

  // Hoist the per-lane WMMA A-operand (Gaussian band matrix) out of the map
  // loop: G[r][j] = g[j-r] for j-r in [0,10], else 0. Built once via LDS
  // (dynamic index -> one ds_load, not a register select chain).
  v2f acoef[7];
#pragma unroll
  for (int c = 0; c < 7; ++c) {
    int j0 = 4 * c + kb;
    int d0 = j0 - M, d1 = j0 + 1 - M;
    acoef[c].x = (d0 >= 0 && d0 <= 10) ? s_gw[d0] : 0.0f;
    acoef[c].y = (d1 >= 0 && d1 <= 10) ? s_gw[d1] : 0.0f;
  }

  const float inv3 = 1.0f / 3.0f;
  float* rawA = s_rawA[wv];
  float* rawB = s_rawB[wv];
  float* rawC = s_rawC[wv];

  // ---- X maps: rawA = mean_c(out), rawB = mean_c(out^2), zero-padded halo.
  for (int cidx = lane; cidx < REG * REG; cidx += 32) {
    int j = cidx / REG, x = cidx % REG;
    int gy = row0 - 5 + j, gx = col0 - 5 + x;
    float o0 = 0.f, o1 = 0.f, o2 = 0.f;
    if (gy >= 0 && gy < H_IMG && gx >= 0 && gx < W_IMG) {
      size_t p = (size_t)gy * W_IMG + gx;
      o0 = out_img[p]; o1 = out_img[NPIX + p]; o2 = out_img[2 * (size_t)NPIX + p];
    }
    rawA[j * RSTR + x] = (o0 + o1 + o2) * inv3;
    rawB[j * RSTR + x] = (o0 * o0 + o1 * o1 + o2 * o2) * inv3;
  }
  __syncthreads();
  v8f muX = blur_tile(rawA, gw, acoef, n, kb);
  v8f sigX = blur_tile(rawB, gw, acoef, n, kb);
#pragma unroll
  for (int i = 0; i < 8; ++i) sigX[i] = sigX[i] - muX[i] * muX[i];

  float bestSig[8], bestCs[8];
  int bestIdx[8];
#pragma unroll
  for (int i = 0; i < 8; ++i) { bestSig[i] = -3.4e38f; bestCs[i] = 0.f; bestIdx[i] = 0; }

  for (int k = 0; k < K_EXP; ++k) {
    __syncthreads();  // previous consumers of raw buffers are done
    const float* bk = bunch + (size_t)k * 3 * NPIX;
    for (int cidx = lane; cidx < REG * REG; cidx += 32) {
      int j = cidx / REG, x = cidx % REG;
      int gy = row0 - 5 + j, gx = col0 - 5 + x;
      float b0 = 0.f, b1 = 0.f, b2 = 0.f, o0 = 0.f, o1 = 0.f, o2 = 0.f;
      if (gy >= 0 && gy < H_IMG && gx >= 0 && gx < W_IMG) {
        size_t p = (size_t)gy * W_IMG + gx;
        b0 = bk[p]; b1 = bk[NPIX + p]; b2 = bk[2 * (size_t)NPIX + p];
        o0 = out_img[p]; o1 = out_img[NPIX + p]; o2 = out_img[2 * (size_t)NPIX + p];
      }
      rawA[j * RSTR + x] = (b0 + b1 + b2) * inv3;
      rawB[j * RSTR + x] = (b0 * b0 + b1 * b1 + b2 * b2) * inv3;
      rawC[j * RSTR + x] = (o0 * b0 + o1 * b1 + o2 * b2) * inv3;
    }
    __syncthreads();
    v8f muY = blur_tile(rawA, gw, acoef, n, kb);
    v8f sYY = blur_tile(rawB, gw, acoef, n, kb);
    v8f sXY = blur_tile(rawC, gw, acoef, n, kb);

    float msum = 0.0f;
#pragma unroll
    for (int i = 0; i < 8; ++i) {
      float m = muY[i];
      float sigY  = sYY[i] - m * m;
      float sigXY = sXY[i] - muX[i] * m;
      float cs = (2.0f * sigXY + C2_V) / fmaxf(sigX[i] + sigY + C2_V, 1e-6f);
      if (sigY > bestSig[i]) { bestSig[i] = sigY; bestCs[i] = cs; bestIdx[i] = k; }
      msum += m;
      int Mrow = i + ((lane >> 4) << 3);
      size_t p = (size_t)(row0 + Mrow) * W_IMG + (size_t)(col0 + n);
      muY_maps[(size_t)k * NPIX + p] = m;
    }
    // wave reduction of the tile's muY sum (deterministic; no atomics)
#pragma unroll
    for (int off = 16; off > 0; off >>= 1) msum += __shfl_xor(msum, off, 32);
    if (lane == 0) tileSums[k * NTILES + tile] = msum;
  }

#pragma unroll
  for (int i = 0; i < 8; ++i) {
    int Mrow = i + ((lane >> 4) << 3);
    size_t p = (size_t)(row0 + Mrow) * W_IMG + (size_t)(col0 + n);
    muX_map[p] = muX[i];
    cs_map[p]  = bestCs[i];
    patch_index[p] = bestIdx[i];
  }
}

__global__ __launch_bounds__(256) void mef_phase2_lg(
    const float* __restrict__ tileSums, float* __restrict__ lG)
{
  __shared__ float red[256];
  const int tid = threadIdx.x;
  for (int k = 0; k < K_EXP; ++k) {
    float s = 0.0f;
    for (int t = tid; t < NTILES; t += 256) s += tileSums[k * NTILES + t];
    red[tid] = s;
    __syncthreads();
    for (int off = 128; off > 0; off >>= 1) {
      if (tid < off) red[tid] += red[tid + off];
      __syncthreads();
    }
    if (tid == 0) {
      float lY = red[0] / (float)NPIX;
      float d = lY - 0.5f;
      lG[k] = expf(-(d * d) / DENOM);
    }
    __syncthreads();
  }
}

__global__ __launch_bounds__(256) void mef_phase3(
    const float* __restrict__ muY_maps, const float* __restrict__ muX_map,
    const float* __restrict__ cs_map, const float* __restrict__ lG,
    float* __restrict__ partials)
{
  __shared__ float red[256];
  const int tid = threadIdx.x;
  const size_t p = (size_t)blockIdx.x * 256 + tid;

  float lg[8];
#pragma unroll
  for (int k = 0; k < 8; ++k) lg[k] = lG[k];

  float num = 0.0f, den = 0.0f;
#pragma unroll
  for (int k = 0; k < 8; ++k) {
    float m = muY_maps[(size_t)k * NPIX + p];
    float d = m - 0.5f;
    float w = lg[k] * expf(-(d * d) / DENOM);
    num += w * m;
    den += w;
  }
  float muYv = num / den;
  float mx = muX_map[p];
  float l = (2.0f * mx * muYv + C1_V) / fmaxf(mx * mx + muYv * muYv + C1_V, 1e-6f);
  red[tid] = l * cs_map[p];
  __syncthreads();
  for (int off = 128; off > 0; off >>= 1) {
    if (tid < off) red[tid] += red[tid + off];
    __syncthreads();
  }
  if (tid == 0) partials[blockIdx.x] = red[0];
}

__global__ __launch_bounds__(256) void mef_phase4(
    const float* __restrict__ partials, float* __restrict__ q)
{
  __shared__ float red[256];
  const int tid = threadIdx.x;
  float s = 0.0f;
  for (int t = tid; t < NPIX / 256; t += 256) s += partials[t];
  red[tid] = s;
  __syncthreads();
  for (int off = 128; off > 0; off >>= 1) {
    if (tid < off) red[tid] += red[tid + off];
    __syncthreads();
  }
  if (tid == 0) q[0] = red[0] / (float)NPIX;
}

extern "C" void kernel_launch(void* const* d_in, const int* in_sizes, int n_in,
                              void* d_out, int out_size, void* d_ws, size_t ws_size,
                              hipStream_t stream) {
  const float* out_img = (const float*)d_in[0];  // [1,3,1024,1024]
  const float* bunch   = (const float*)d_in[1];  // [8,3,1024,1024]
  const float* window  = (const float*)d_in[2];  // [1,3,11,11]

  float* ws = (float*)d_ws;
  float* muY_maps = ws;                              // 8*NPIX
  float* muX_map  = muY_maps + (size_t)8 * NPIX;     // NPIX
  float* cs_map   = muX_map + NPIX;                  // NPIX
  float* tileSums = cs_map + NPIX;                   // 8*NTILES
  float* lG       = tileSums + 8 * NTILES;           // 8
  float* partials = lG + 8;                          // NPIX/256

  float* qout = (float*)d_out;
  int*   pidx = ((int*)d_out) + 1;

  mef_phase1<<<NTILES / WAVES, 256, 0, stream>>>(out_img, bunch, window,
                                                 muY_maps, muX_map, cs_map,
                                                 tileSums, pidx);
  mef_phase2_lg<<<1, 256, 0, stream>>>(tileSums, lG);
  mef_phase3<<<NPIX / 256, 256, 0, stream>>>(muY_maps, muX_map, cs_map, lG, partials);
  mef_phase4<<<1, 256, 0, stream>>>(partials, qout);
}